// GCN_43765716746406
// MI455X (gfx1250) — compile-verified
//
#include <hip/hip_runtime.h>

#define N_NODES 50000
#define N_EDGES 800000
#define D_MODEL 128
#define N_LAYERS 3

typedef __attribute__((ext_vector_type(2))) float v2f;
typedef __attribute__((ext_vector_type(8))) float v8f;

// ---------------------------------------------------------------------------
// Degree accumulation: one thread per edge, two f32 atomics.
// ---------------------------------------------------------------------------
__global__ __launch_bounds__(256) void gcn_degrees(const int* __restrict__ src,
                                                   const int* __restrict__ dst,
                                                   float* __restrict__ out_deg,
                                                   float* __restrict__ in_deg) {
  int e = blockIdx.x * 256 + threadIdx.x;
  if (e < N_EDGES) {
    atomicAdd(&out_deg[src[e]], 1.0f);
    atomicAdd(&in_deg[dst[e]], 1.0f);
  }
}

// In-place deg -> rsqrt(max(deg,1))
__global__ __launch_bounds__(256) void gcn_norms(float* __restrict__ out_n,
                                                 float* __restrict__ in_n) {
  int i = blockIdx.x * 256 + threadIdx.x;
  if (i < N_NODES) {
    out_n[i] = rsqrtf(fmaxf(out_n[i], 1.0f));
    in_n[i]  = rsqrtf(fmaxf(in_n[i], 1.0f));
  }
}

// ---------------------------------------------------------------------------
// Edge scatter: one wave32 per edge; each lane moves a float4 (32*4 = 128).
// agg[dst] += h[src] * out_norm[src]   via global_atomic_add_f32 (no return).
// ---------------------------------------------------------------------------
__global__ __launch_bounds__(256) void gcn_scatter(const float* __restrict__ h,
                                                   const int* __restrict__ src,
                                                   const int* __restrict__ dst,
                                                   const float* __restrict__ out_norm,
                                                   float* __restrict__ agg) {
  long long gid = (long long)blockIdx.x * 256 + threadIdx.x;
  int e = (int)(gid >> 5);
  if (e >= N_EDGES) return;
  int lane = (int)(gid & 31);
  int s = src[e];
  int d = dst[e];
  float nrm = out_norm[s];
  const float4* hp = (const float4*)(h + (size_t)s * D_MODEL);
  float4 v = hp[lane];
  float* ap = agg + (size_t)d * D_MODEL + lane * 4;
  atomicAdd(ap + 0, v.x * nrm);
  atomicAdd(ap + 1, v.y * nrm);
  atomicAdd(ap + 2, v.z * nrm);
  atomicAdd(ap + 3, v.w * nrm);
}

// ---------------------------------------------------------------------------
// Fused GEMM: out[16 rows x 128] = (in_norm[row] * agg[row]) @ W + bias
// 128 threads = 4 waves; wave w covers columns [32w, 32w+32) as two 16x16
// WMMA tiles. K=128 consumed as 32 x V_WMMA_F32_16X16X4_F32 (exact f32).
// W staged through LDS in two 64-row halves (rows padded to 136 floats).
// ---------------------------------------------------------------------------
__global__ __launch_bounds__(128) void gcn_gemm(const float* __restrict__ agg,
                                                const float* __restrict__ in_norm,
                                                const float* __restrict__ W,
                                                const float* __restrict__ bias,
                                                float* __restrict__ out) {
  __shared__ float As[16][D_MODEL + 4];   //  8.4 KB, +4 pad to spread banks
  __shared__ float Bs[64][D_MODEL + 8];   // 34.8 KB, +8 pad: 32 distinct banks

  const int tid = threadIdx.x;
  const int wave = tid >> 5;
  const int lane = tid & 31;
  const int lrow = lane & 15;              // M (A frag) / N (B frag) lane index
  const int koff = (lane >= 16) ? 2 : 0;   // high half-wave holds K+2, K+3
  const int row0 = blockIdx.x * 16;
  const int nb0 = wave * 32;               // wave's column base

  // ---- Stage A tile (scaled by in_norm); zero-fill past N_NODES ----
  {
    int r = tid >> 3;              // 0..15
    int c0 = (tid & 7) * 16;       // 0,16,...,112
    int grow = row0 + r;
    if (grow < N_NODES) {
      float nrm = in_norm[grow];
      const float4* s4 = (const float4*)(agg + (size_t)grow * D_MODEL + c0);
#pragma unroll
      for (int j = 0; j < 4; ++j) {
        float4 v = s4[j];
        As[r][c0 + 4 * j + 0] = v.x * nrm;
        As[r][c0 + 4 * j + 1] = v.y * nrm;
        As[r][c0 + 4 * j + 2] = v.z * nrm;
        As[r][c0 + 4 * j + 3] = v.w * nrm;
      }
    } else {
#pragma unroll
      for (int j = 0; j < 16; ++j) As[r][c0 + j] = 0.0f;
    }
  }

  v8f acc0 = {};
  v8f acc1 = {};

  for (int half = 0; half < 2; ++half) {
    __syncthreads();  // A ready (iter 0); previous Bs reads done (iter 1)
    // ---- Stage 64 rows of W: 2048 float4, 16 per thread, coalesced ----
    {
      const float4* wsrc = (const float4*)(W + (size_t)half * 64 * D_MODEL);
      for (int i = tid; i < 2048; i += 128) {
        float4 v = wsrc[i];
        int r = i >> 5;            // 32 float4 per row
        int c = (i & 31) * 4;
        Bs[r][c + 0] = v.x;
        Bs[r][c + 1] = v.y;
        Bs[r][c + 2] = v.z;
        Bs[r][c + 3] = v.w;
      }
    }
    __syncthreads();

#pragma unroll
    for (int kk = 0; kk < 16; ++kk) {
      const int kg = half * 64 + kk * 4 + koff;  // global K for A
      const int kl = kk * 4 + koff;              // local K for staged W
      v2f a, b0, b1;
      a.x = As[lrow][kg];
      a.y = As[lrow][kg + 1];
      b0.x = Bs[kl][nb0 + lrow];
      b0.y = Bs[kl + 1][nb0 + lrow];
      b1.x = Bs[kl][nb0 + 16 + lrow];
      b1.y = Bs[kl + 1][nb0 + 16 + lrow];
      acc0 = __builtin_amdgcn_wmma_f32_16x16x4_f32(false, a, false, b0,
                                                   (short)0, acc0, false, false);
      acc1 = __builtin_amdgcn_wmma_f32_16x16x4_f32(false, a, false, b1,
                                                   (short)0, acc1, false, false);
    }
  }

  // ---- Store D + bias: VGPR r -> row (lane<16 ? r : 8+r), col = nb + lrow ----
  const int mbase = (lane < 16) ? 0 : 8;
  const int col0 = nb0 + lrow;
  const float bia0 = bias[col0];
  const float bia1 = bias[col0 + 16];
#pragma unroll
  for (int r = 0; r < 8; ++r) {
    int grow = row0 + mbase + r;
    if (grow < N_NODES) {
      out[(size_t)grow * D_MODEL + col0] = acc0[r] + bia0;
      out[(size_t)grow * D_MODEL + col0 + 16] = acc1[r] + bia1;
    }
  }
}

// ---------------------------------------------------------------------------
// Host side
// ---------------------------------------------------------------------------
static inline size_t align_up(size_t x, size_t a) { return (x + a - 1) & ~(a - 1); }

extern "C" void kernel_launch(void* const* d_in, const int* in_sizes, int n_in,
                              void* d_out, int out_size, void* d_ws, size_t ws_size,
                              hipStream_t stream) {
  (void)in_sizes; (void)n_in; (void)out_size; (void)ws_size;

  const float* in_feat = (const float*)d_in[0];
  const int* ei = (const int*)d_in[1];
  const int* src = ei;
  const int* dst = ei + N_EDGES;
  const float* Ws = (const float*)d_in[2];   // [3][128][128]
  const float* bs = (const float*)d_in[3];   // [3][128]
  float* out = (float*)d_out;

  // Workspace carve-up (256B aligned regions)
  char* ws = (char*)d_ws;
  size_t off = 0;
  float* out_norm = (float*)(ws + off); off = align_up(off + N_NODES * sizeof(float), 256);
  float* in_norm  = (float*)(ws + off); off = align_up(off + N_NODES * sizeof(float), 256);
  float* agg      = (float*)(ws + off); off = align_up(off + (size_t)N_NODES * D_MODEL * sizeof(float), 256);
  float* hbuf     = (float*)(ws + off); off = align_up(off + (size_t)N_NODES * D_MODEL * sizeof(float), 256);

  // Degrees -> norms (recomputed every call; deterministic)
  hipMemsetAsync(out_norm, 0, N_NODES * sizeof(float), stream);
  hipMemsetAsync(in_norm, 0, N_NODES * sizeof(float), stream);
  gcn_degrees<<<(N_EDGES + 255) / 256, 256, 0, stream>>>(src, dst, out_norm, in_norm);
  gcn_norms<<<(N_NODES + 255) / 256, 256, 0, stream>>>(out_norm, in_norm);

  const float* h = in_feat;
  for (int l = 0; l < N_LAYERS; ++l) {
    hipMemsetAsync(agg, 0, (size_t)N_NODES * D_MODEL * sizeof(float), stream);
    long long sthreads = (long long)N_EDGES * 32;
    int sblocks = (int)((sthreads + 255) / 256);
    gcn_scatter<<<sblocks, 256, 0, stream>>>(h, src, dst, out_norm, agg);
    float* o = (l == N_LAYERS - 1) ? out : hbuf;
    gcn_gemm<<<(N_NODES + 15) / 16, 128, 0, stream>>>(
        agg, in_norm, Ws + (size_t)l * D_MODEL * D_MODEL, bs + (size_t)l * D_MODEL, o);
    h = o;
  }
}